// AggrSum_59322088292862
// MI455X (gfx1250) — compile-verified
//
#include <hip/hip_runtime.h>
#include <hip/hip_bf16.h>
#include <stdint.h>

// Segment-sum (one-hot matmul == scatter/gather sum of H rows keyed by X_neis).
// N=32768 rows, D=256 cols, V=8192 buckets. Memory-bound (~40MB); fits in L2.
//
// Pipeline:
//   k0: zero counts[V]
//   k1: counts[v]++                      (global_atomic_add_u32)
//   k2: exclusive scan -> offsets[V+1], cursor[V]
//   k3: CSR fill: perm[cursor[v]++] = n  (atomics; order fixed up in k4)
//   k4: per-bucket gather-sum, rows staged to LDS via async copy engine
//       (global_load_async_to_lds_b128 + s_wait_asynccnt), accumulation in
//       ascending-n order => deterministic fp result.

#define AGS_D      256          // row length (floats); 1KB per row
#define AGS_DEPTH  8            // async pipeline depth (rows staged per chunk)

__global__ void ags_zero_u32(uint32_t* p, int n) {
    int i = blockIdx.x * blockDim.x + threadIdx.x;
    if (i < n) p[i] = 0u;
}

__global__ void ags_count(const int* __restrict__ X, uint32_t* __restrict__ counts, int N) {
    int n = blockIdx.x * blockDim.x + threadIdx.x;
    if (n < N) {
        int v = X[n];
        __hip_atomic_fetch_add(&counts[v], 1u, __ATOMIC_RELAXED, __HIP_MEMORY_SCOPE_AGENT);
    }
}

// Single-block exclusive scan over counts[V] (V <= 8192, blockDim.x == 1024).
__global__ void ags_scan(const uint32_t* __restrict__ counts,
                         uint32_t* __restrict__ offsets,
                         uint32_t* __restrict__ cursor, int V) {
    __shared__ uint32_t part[1024];
    const int t = threadIdx.x;
    const int ITEMS = 8;                       // 8192 / 1024
    uint32_t c[ITEMS];
    uint32_t s = 0;
    for (int i = 0; i < ITEMS; ++i) {
        int idx = t * ITEMS + i;
        c[i] = (idx < V) ? counts[idx] : 0u;
        s += c[i];
    }
    part[t] = s;
    __syncthreads();
    // Hillis-Steele inclusive scan over 1024 partials
    for (int off = 1; off < 1024; off <<= 1) {
        uint32_t v = (t >= off) ? part[t - off] : 0u;
        __syncthreads();
        part[t] += v;
        __syncthreads();
    }
    uint32_t excl = part[t] - s;               // exclusive prefix for this thread's run
    for (int i = 0; i < ITEMS; ++i) {
        int idx = t * ITEMS + i;
        if (idx < V) { offsets[idx] = excl; cursor[idx] = excl; }
        excl += c[i];
    }
    if (t == 1023) offsets[V] = part[1023];
}

__global__ void ags_fill(const int* __restrict__ X, uint32_t* __restrict__ cursor,
                         uint32_t* __restrict__ perm, int N) {
    int n = blockIdx.x * blockDim.x + threadIdx.x;
    if (n < N) {
        int v = X[n];
        uint32_t pos = __hip_atomic_fetch_add(&cursor[v], 1u, __ATOMIC_RELAXED,
                                              __HIP_MEMORY_SCOPE_AGENT);
        perm[pos] = (uint32_t)n;
    }
}

// One 64-thread block (2 wave32) per output bucket v.
// Wave w owns columns [w*128, w*128+128): lane handles 4 floats (16B).
// Each wave stages its 512B half-row slices into wave-private LDS slots with
// the async copy engine, waits on its own ASYNCcnt (no cross-wave barriers,
// so the data-dependent loop is divergence-safe), then accumulates.
__global__ void ags_segment_sum(const float* __restrict__ H,
                                const uint32_t* __restrict__ perm,
                                const uint32_t* __restrict__ offsets,
                                float* __restrict__ out) {
    __shared__ __align__(16) float stage[2][AGS_DEPTH][128];  // 8KB

    const int v    = blockIdx.x;
    const int wave = threadIdx.x >> 5;
    const int lane = threadIdx.x & 31;

    const int beg = (int)offsets[v];
    const int end = (int)offsets[v + 1];
    const int len = end - beg;

    float4 acc; acc.x = 0.f; acc.y = 0.f; acc.z = 0.f; acc.w = 0.f;

    int last = -1;   // select segment members in ascending n => deterministic fp order
    int done = 0;
    while (done < len) {
        int m = (len - done) < AGS_DEPTH ? (len - done) : AGS_DEPTH;
        // Make sure prior ds reads of the stage buffer retired before the
        // async engine overwrites it.
        asm volatile("s_wait_dscnt 0x0" ::: "memory");
        for (int i = 0; i < m; ++i) {
            // O(k^2) uniform selection of the next-smallest row index (k ~ 4).
            int cur = 0x7fffffff;
            for (int j = beg; j < end; ++j) {
                int n = (int)perm[j];
                if (n > last && n < cur) cur = n;
            }
            last = cur;
            // global byte offset: n*D*4 + wave*512 + lane*16   (D==256 -> n<<10)
            uint32_t goff = ((uint32_t)cur << 10) + ((uint32_t)wave << 9) +
                            ((uint32_t)lane << 4);
            // LDS byte offset of this lane's 16B slot (low 32 bits of the
            // generic pointer == LDS offset on gfx1250).
            uint32_t loff = (uint32_t)(uintptr_t)(void*)&stage[wave][i][lane << 2];
            asm volatile("global_load_async_to_lds_b128 %0, %1, %2"
                         :: "v"(loff), "v"(goff), "s"(H)
                         : "memory");
        }
        asm volatile("s_wait_asynccnt 0x0" ::: "memory");
        for (int i = 0; i < m; ++i) {
            const float4 h = *reinterpret_cast<const float4*>(&stage[wave][i][lane << 2]);
            acc.x += h.x; acc.y += h.y; acc.z += h.z; acc.w += h.w;
        }
        done += m;
    }

    const size_t o = (size_t)v * AGS_D + ((size_t)wave << 7) + ((size_t)lane << 2);
    *reinterpret_cast<float4*>(out + o) = acc;   // empty buckets write zeros
}

extern "C" void kernel_launch(void* const* d_in, const int* in_sizes, int n_in,
                              void* d_out, int out_size, void* d_ws, size_t ws_size,
                              hipStream_t stream) {
    const float* H = (const float*)d_in[0];
    const int*   X = (const int*)d_in[1];   // indices in [0, V)
    float*     out = (float*)d_out;

    const int N = in_sizes[1];              // 32768
    const int D = in_sizes[0] / N;          // 256 (kernel assumes 256)
    const int V = out_size / D;             // 8192
    (void)n_in; (void)ws_size;

    uint32_t* counts  = (uint32_t*)d_ws;          // V
    uint32_t* offsets = counts + V;               // V + 1
    uint32_t* cursor  = offsets + V + 1;          // V
    uint32_t* perm    = cursor + V;               // N   (~229 KB total)

    ags_zero_u32<<<(V + 255) / 256, 256, 0, stream>>>(counts, V);
    ags_count   <<<(N + 255) / 256, 256, 0, stream>>>(X, counts, N);
    ags_scan    <<<1, 1024, 0, stream>>>(counts, offsets, cursor, V);
    ags_fill    <<<(N + 255) / 256, 256, 0, stream>>>(X, cursor, perm, N);
    ags_segment_sum<<<V, 64, 0, stream>>>(H, perm, offsets, out);
}